// GNNAtomic_20366734917670
// MI455X (gfx1250) — compile-verified
//
#include <hip/hip_runtime.h>

typedef float v2f __attribute__((ext_vector_type(2)));
typedef float v8f __attribute__((ext_vector_type(8)));

#define BS   8
#define NN   1024
#define JJ   3
#define FIN  32
#define NIN  96      // J * F_in
#define FOUT 64
#define TILE_N 64    // output rows per block
#define MCH  32      // K-chunk (m) per LDS stage

// Phase-1 LDS (main GEMM), double-buffered
#define PW_WW 100    // [row][m*3+j]; mult of 4 (b128 stores); 100%64=36, gcd(36,64)=4 -> 16 distinct banks; 36d==6 mod 64 unsolvable -> halves disjoint
#define PW_X2 96     // [m/2][f*2+p]; 96%64=32 -> upper half uses the other 32 banks
// Phase-2 LDS (fused FC), aliases phase-1
#define PW_W2 160    // [k/2][col*2+p]; 160%64=32 -> halves disjoint
#define PW_Y  98     // [row][k]; even -> b64-aligned A reads

#define WW_DW (TILE_N * PW_WW)            // 6400
#define X2_DW ((MCH / 2) * PW_X2)         // 1536
#define BUF_DW (WW_DW + X2_DW)            // 7936 per buffer
#define W2_DW (( NIN / 2) * PW_W2)        // 7680
#define Y_DW  (TILE_N * PW_Y)             // 6272
#define SM_DW (2 * BUF_DW)                // 15872 dwords = 63.5 KB (phase2 needs 13952)

static __device__ __forceinline__ v8f wmma4(v2f a, v2f b, v8f c) {
  return __builtin_amdgcn_wmma_f32_16x16x4_f32(false, a, false, b,
                                               (short)0, c, false, false);
}

// ---------------- Kernel A: einsum GEMM + fused FC1/FC2 (WMMA f32) ----------
__global__ __launch_bounds__(192) void gemm_fc_kernel(
    const float* __restrict__ WW, const float* __restrict__ X,
    const float* __restrict__ W1, const float* __restrict__ B1,
    const float* __restrict__ W2, const float* __restrict__ B2,
    float* __restrict__ Z)
{
  __shared__ __align__(16) float smem[SM_DW];
  float* w2 = smem;             // phase 2: k-paired fused FC weights
  float* yl = smem + W2_DW;     // phase 2: Y tile (64 x 96)

  const int tid  = threadIdx.x;
  const int b    = blockIdx.x >> 4;            // 8 batches
  const int n0   = (blockIdx.x & 15) * TILE_N; // 16 row-tiles per batch
  const int w    = tid >> 5;                   // wave 0..5
  const int lane = tid & 31;
  const int half = lane >> 4;                  // K-half (K0/K1 vs K2/K3)
  const int lr   = lane & 15;
  const int j    = w % 3;                      // adjacency slice
  const int rh   = w / 3;                      // which 32-row half

  v8f acc00 = {}, acc01 = {}, acc10 = {}, acc11 = {};  // [row-tile][f-tile]

  // WW tile copy mapping: 192 threads = 64 rows x 3 parts of 32 floats
  const int cprow  = tid / 3;
  const int cppart = tid % 3;
  const float* wsrc = WW + (size_t)(b * NN + n0 + cprow) * NN * JJ + cppart * 32;
  // x tile copy mapping (first 128 threads): 16 m-pairs x 8 f-quads
  const int m2c = tid >> 3;
  const int fqc = (tid & 7) * 4;

  float4 rw[8];           // staged WW chunk (b128 global loads)
  float4 rx0, rx1;        // staged x rows

  auto globalLoad = [&](int m0) {
    const float4* s = (const float4*)(wsrc + (size_t)m0 * JJ);
    #pragma unroll
    for (int i = 0; i < 8; ++i) rw[i] = s[i];
    if (tid < 128) {
      const float* xs = X + (size_t)(b * NN + m0 + 2 * m2c) * FIN + fqc;
      rx0 = *(const float4*)xs;
      rx1 = *(const float4*)(xs + FIN);
    }
  };
  auto ldsStore = [&](int p) {
    float* wd = smem + p * BUF_DW + cprow * PW_WW + cppart * 32;
    #pragma unroll
    for (int i = 0; i < 8; ++i) ((float4*)wd)[i] = rw[i];
    if (tid < 128) {
      float2* d = (float2*)(smem + p * BUF_DW + WW_DW + m2c * PW_X2 + fqc * 2);
      d[0] = make_float2(rx0.x, rx1.x);
      d[1] = make_float2(rx0.y, rx1.y);
      d[2] = make_float2(rx0.z, rx1.z);
      d[3] = make_float2(rx0.w, rx1.w);
    }
  };

  // ---- software pipeline: prime buffer 0
  globalLoad(0);
  ldsStore(0);
  __syncthreads();

  int p = 0;
  for (int m0 = 0; m0 < NN; m0 += MCH) {
    const bool more = (m0 + MCH) < NN;
    if (more) globalLoad(m0 + MCH);          // overlap HBM with WMMAs below

    const float* ww = smem + p * BUF_DW;
    const float* x2 = ww + WW_DW;
    #pragma unroll
    for (int k4 = 0; k4 < MCH; k4 += 4) {    // 8 K-steps of 4
      const int km = k4 + 2 * half;
      const int r0 = rh * 32 + lr;
      v2f a0, a1, bf0, bf1;
      a0[0] = ww[r0 * PW_WW + km * 3 + j];
      a0[1] = ww[r0 * PW_WW + km * 3 + 3 + j];
      a1[0] = ww[(r0 + 16) * PW_WW + km * 3 + j];
      a1[1] = ww[(r0 + 16) * PW_WW + km * 3 + 3 + j];
      bf0 = *(const v2f*)&x2[(km >> 1) * PW_X2 + lr * 2];
      bf1 = *(const v2f*)&x2[(km >> 1) * PW_X2 + (16 + lr) * 2];
      acc00 = wmma4(a0, bf0, acc00);
      acc01 = wmma4(a0, bf1, acc01);
      acc10 = wmma4(a1, bf0, acc10);
      acc11 = wmma4(a1, bf1, acc11);
    }

    if (more) ldsStore(p ^ 1);               // commit next chunk to other buffer
    __syncthreads();
    p ^= 1;
  }
  // final in-loop barrier: all waves done reading phase-1 buffers

  // ---- spill Y tile (64 x 96) and load k-paired FC weights
  #pragma unroll
  for (int v = 0; v < 8; ++v) {
    const int rl = half ? v + 8 : v;
    const int r0 = rh * 32 + rl;
    yl[r0 * PW_Y + j * 32 + lr]             = acc00[v];
    yl[r0 * PW_Y + j * 32 + 16 + lr]        = acc01[v];
    yl[(r0 + 16) * PW_Y + j * 32 + lr]      = acc10[v];
    yl[(r0 + 16) * PW_Y + j * 32 + 16 + lr] = acc11[v];
  }
  for (int l = tid; l < FOUT * NIN; l += 192) {
    const int col = l / NIN, k = l % NIN;
    const float v = (col < 32) ? W1[col * NIN + k] : W2[(col - 32) * NIN + k];
    w2[(k >> 1) * PW_W2 + col * 2 + (k & 1)] = v;
  }
  __syncthreads();

  // ---- fused FC: 4 waves x (16 rows, 64 cols); all-b64 operand feeds
  if (w < 4) {
    v8f c0 = {}, c1 = {}, c2 = {}, c3 = {};
    const int arow = w * 16 + lr;
    #pragma unroll
    for (int kk = 0; kk < NIN; kk += 4) {
      v2f a = *(const v2f*)&yl[arow * PW_Y + kk + 2 * half];
      const float* wrow = &w2[((kk >> 1) + half) * PW_W2];
      v2f q0 = *(const v2f*)&wrow[lr * 2];
      v2f q1 = *(const v2f*)&wrow[(16 + lr) * 2];
      v2f q2 = *(const v2f*)&wrow[(32 + lr) * 2];
      v2f q3 = *(const v2f*)&wrow[(48 + lr) * 2];
      c0 = wmma4(a, q0, c0);
      c1 = wmma4(a, q1, c1);
      c2 = wmma4(a, q2, c2);
      c3 = wmma4(a, q3, c3);
    }
    const float bs0 = B1[lr], bs1 = B1[16 + lr];
    const float bs2 = B2[lr], bs3 = B2[16 + lr];
    #pragma unroll
    for (int v = 0; v < 8; ++v) {
      const int rl = half ? v + 8 : v;
      const size_t grow = (size_t)(b * NN + n0 + w * 16 + rl);
      Z[grow * FOUT + lr]      = fmaxf(c0[v] + bs0, 0.f);
      Z[grow * FOUT + 16 + lr] = fmaxf(c1[v] + bs1, 0.f);
      Z[grow * FOUT + 32 + lr] = c2[v] + bs2;
      Z[grow * FOUT + 48 + lr] = c3[v] + bs3;
    }
  }
}

// ---------------- Kernel B: deterministic per-column mean / rsqrt(var+eps) ---
__global__ __launch_bounds__(256) void stats_kernel(const float* __restrict__ Z,
                                                    float* __restrict__ stats)
{
  __shared__ float s_sum[256];
  __shared__ float s_sq[256];
  const int c = blockIdx.x;
  const int t = threadIdx.x;
  float s = 0.f, q = 0.f;
  for (int r = t; r < BS * NN; r += 256) {
    const float v = Z[(size_t)r * FOUT + c];
    s += v; q += v * v;
  }
  s_sum[t] = s; s_sq[t] = q;
  __syncthreads();
  for (int off = 128; off > 0; off >>= 1) {
    if (t < off) { s_sum[t] += s_sum[t + off]; s_sq[t] += s_sq[t + off]; }
    __syncthreads();
  }
  if (t == 0) {
    const float inv_n = 1.0f / (float)(BS * NN);
    const float mean = s_sum[0] * inv_n;
    const float var  = s_sq[0] * inv_n - mean * mean;   // biased, matches jnp.var
    stats[c * 2]     = mean;
    stats[c * 2 + 1] = rsqrtf(var + 1e-5f);
  }
}

// ---------------- Kernel C: apply BN affine --------------------------------
__global__ __launch_bounds__(256) void bn_kernel(const float* __restrict__ Z,
    const float* __restrict__ stats, const float* __restrict__ gamma,
    const float* __restrict__ beta, float* __restrict__ out)
{
  const int i = blockIdx.x * 256 + threadIdx.x;
  const int c = i & (FOUT - 1);
  out[i] = (Z[i] - stats[c * 2]) * stats[c * 2 + 1] * gamma[c] + beta[c];
}

extern "C" void kernel_launch(void* const* d_in, const int* in_sizes, int n_in,
                              void* d_out, int out_size, void* d_ws, size_t ws_size,
                              hipStream_t stream) {
  const float* WW = (const float*)d_in[0];
  const float* X  = (const float*)d_in[1];
  const float* W1 = (const float*)d_in[2];
  const float* B1 = (const float*)d_in[3];
  const float* W2 = (const float*)d_in[4];
  const float* B2 = (const float*)d_in[5];
  const float* gm = (const float*)d_in[6];
  const float* bt = (const float*)d_in[7];
  (void)in_sizes; (void)n_in; (void)out_size; (void)ws_size;

  float* Z     = (float*)d_ws;                 // 8192 x 64 f32 = 2 MB
  float* stats = Z + (size_t)BS * NN * FOUT;   // 128 f32

  gemm_fc_kernel<<<BS * (NN / TILE_N), 192, 0, stream>>>(WW, X, W1, B1, W2, B2, Z);
  stats_kernel<<<FOUT, 256, 0, stream>>>(Z, stats);
  bn_kernel<<<(BS * NN * FOUT) / 256, 256, 0, stream>>>(Z, stats, gm, bt, (float*)d_out);
}